// StaticGNN_5351529251150
// MI455X (gfx1250) — compile-verified
//
#include <hip/hip_runtime.h>
#include <hip/hip_bf16.h>
#include <math.h>

typedef float v2f __attribute__((ext_vector_type(2)));
typedef float v8f __attribute__((ext_vector_type(8)));

#define D 128
#define NEG_SLOPE 0.2f

__device__ __forceinline__ float lrelu(float x) {
    return x > 0.0f ? x : NEG_SLOPE * x;
}

// ---------------------------------------------------------------------------
// H[N,128] = act(X[N,128]) @ W[128,128]   (act = relu if relu_in, else id)
// One wave computes a 16x16 tile of H via 32 x V_WMMA_F32_16X16X4_F32.
// A layout (16x4 f32): lane&15 = M row; lane>=16 holds K=2,3 (else K=0,1).
// B layout (4x16 f32): lane&15 = N col; same K-pair split across lane halves.
// C/D layout (16x16 f32): VGPR r holds M=r (lanes 0-15) / M=r+8 (lanes 16-31),
// N = lane&15.
// ---------------------------------------------------------------------------
__global__ __launch_bounds__(256)
void gat_gemm_wmma(const float* __restrict__ X, const float* __restrict__ W,
                   float* __restrict__ H, int N, int relu_in) {
    int wave = (int)((blockIdx.x * blockDim.x + threadIdx.x) >> 5);
    int lane = threadIdx.x & 31;
    const int ntiles_n = D / 16;                 // 8
    int mtile = wave / ntiles_n;
    int ntile = wave - mtile * ntiles_n;
    int m0 = mtile * 16;
    if (m0 >= N) return;
    int n0 = ntile * 16;

    int half = lane >> 4;                        // 0: K=0,1  1: K=2,3
    int l    = lane & 15;                        // M row (A) / N col (B,C,D)

    const float* __restrict__ xrow = X + (size_t)(m0 + l) * D;
    v8f c = {};

    for (int k = 0; k < D; k += 4) {
        float a0 = xrow[k + 2 * half + 0];
        float a1 = xrow[k + 2 * half + 1];
        if (relu_in) { a0 = fmaxf(a0, 0.0f); a1 = fmaxf(a1, 0.0f); }
        v2f a, b;
        a[0] = a0; a[1] = a1;
        b[0] = W[(size_t)(k + 2 * half + 0) * D + n0 + l];
        b[1] = W[(size_t)(k + 2 * half + 1) * D + n0 + l];
        c = __builtin_amdgcn_wmma_f32_16x16x4_f32(
                /*neg_a=*/false, a, /*neg_b=*/false, b,
                /*c_mod=*/(short)0, c, /*reuse_a=*/false, /*reuse_b=*/false);
    }

    float* __restrict__ hrow = H + (size_t)m0 * D + n0 + l;
#pragma unroll
    for (int r = 0; r < 8; ++r) {
        hrow[(size_t)(r + 8 * half) * D] = c[r];
    }
}

// ---------------------------------------------------------------------------
// Per-node attention terms: a_s[i] = dot(h[i], att_s), a_d[i] = dot(h[i], att_d)
// One wave per node; each lane handles 4 contiguous dims, wave32 shfl reduce.
// ---------------------------------------------------------------------------
__global__ __launch_bounds__(256)
void gat_att(const float* __restrict__ H,
             const float* __restrict__ att_s, const float* __restrict__ att_d,
             float* __restrict__ a_s, float* __restrict__ a_d, int N) {
    int wave = (int)((blockIdx.x * blockDim.x + threadIdx.x) >> 5);
    int lane = threadIdx.x & 31;
    if (wave >= N) return;
    const float4* hv4 = (const float4*)(H + (size_t)wave * D);
    const float4* as4 = (const float4*)att_s;
    const float4* ad4 = (const float4*)att_d;
    float4 h = hv4[lane];
    float4 as = as4[lane];
    float4 ad = ad4[lane];
    float ss = h.x * as.x + h.y * as.y + h.z * as.z + h.w * as.w;
    float sd = h.x * ad.x + h.y * ad.y + h.z * ad.z + h.w * ad.w;
#pragma unroll
    for (int off = 16; off > 0; off >>= 1) {
        ss += __shfl_xor(ss, off, 32);
        sd += __shfl_xor(sd, off, 32);
    }
    if (lane == 0) {
        a_s[wave] = ss;
        a_d[wave] = sd;
    }
}

// Self-loop logit + init running max with it (every node has a self-loop).
__global__ __launch_bounds__(256)
void gat_node_init(const float* __restrict__ a_s, const float* __restrict__ a_d,
                   float* __restrict__ eself, float* __restrict__ m, int N) {
    int i = blockIdx.x * blockDim.x + threadIdx.x;
    if (i >= N) return;
    float e = lrelu(a_s[i] + a_d[i]);
    eself[i] = e;
    m[i] = e;
}

// Segment max over incoming edges.
__global__ __launch_bounds__(256)
void gat_edge_max(const int* __restrict__ src, const int* __restrict__ dst,
                  const float* __restrict__ a_s, const float* __restrict__ a_d,
                  float* __restrict__ m, int E) {
    int i = blockIdx.x * blockDim.x + threadIdx.x;
    if (i >= E) return;
    int s = src[i], d = dst[i];
    float e = lrelu(a_s[s] + a_d[d]);
    atomicMax(&m[d], e);
}

// Init segment sum with the self-loop term.
__global__ __launch_bounds__(256)
void gat_node_sum_init(const float* __restrict__ eself, const float* __restrict__ m,
                       float* __restrict__ ssum, int N) {
    int i = blockIdx.x * blockDim.x + threadIdx.x;
    if (i >= N) return;
    ssum[i] = expf(eself[i] - m[i]);
}

// Segment sum of exp(e - m[dst]).
__global__ __launch_bounds__(256)
void gat_edge_sum(const int* __restrict__ src, const int* __restrict__ dst,
                  const float* __restrict__ a_s, const float* __restrict__ a_d,
                  const float* __restrict__ m, float* __restrict__ ssum, int E) {
    int i = blockIdx.x * blockDim.x + threadIdx.x;
    if (i >= E) return;
    int s = src[i], d = dst[i];
    float e = lrelu(a_s[s] + a_d[d]);
    atomicAdd(&ssum[d], expf(e - m[d]));
}

// out[i,:] = bias + alpha_self * h[i,:]   (one thread per element)
__global__ __launch_bounds__(256)
void gat_out_init(const float* __restrict__ H, const float* __restrict__ eself,
                  const float* __restrict__ m, const float* __restrict__ ssum,
                  const float* __restrict__ bias, float* __restrict__ out,
                  long total) {
    long idx = (long)blockIdx.x * blockDim.x + threadIdx.x;
    if (idx >= total) return;
    int node = (int)(idx >> 7);
    int dcol = (int)(idx & (D - 1));
    float w = expf(eself[node] - m[node]) / ssum[node];
    out[idx] = bias[dcol] + w * H[idx];
}

// Weighted aggregation: out[dst,:] += alpha * h[src,:]
// One wave per edge; each lane handles 4 dims (float4 gather, 4 f32 atomics).
__global__ __launch_bounds__(256)
void gat_edge_agg(const float* __restrict__ H,
                  const int* __restrict__ src, const int* __restrict__ dst,
                  const float* __restrict__ a_s, const float* __restrict__ a_d,
                  const float* __restrict__ m, const float* __restrict__ ssum,
                  float* __restrict__ out, int E) {
    int wave = (int)((blockIdx.x * blockDim.x + threadIdx.x) >> 5);
    int lane = threadIdx.x & 31;
    if (wave >= E) return;
    int s = src[wave], d = dst[wave];
    float e = lrelu(a_s[s] + a_d[d]);
    float w = expf(e - m[d]) / ssum[d];
    float4 hv = ((const float4*)(H + (size_t)s * D))[lane];
    float* od = out + (size_t)d * D + lane * 4;
    atomicAdd(od + 0, w * hv.x);
    atomicAdd(od + 1, w * hv.y);
    atomicAdd(od + 2, w * hv.z);
    atomicAdd(od + 3, w * hv.w);
}

// ---------------------------------------------------------------------------

static void run_layer(const float* X, const float* W, const float* att_s_w,
                      const float* att_d_w, const float* bias,
                      const int* src, const int* dst,
                      float* H, float* a_s, float* a_d, float* m, float* ssum,
                      float* eself, float* out, int N, int E, int relu_in,
                      hipStream_t stream) {
    const int ntiles_n = D / 16;                              // 8
    int gemm_waves = (N / 16) * ntiles_n;                     // N % 16 == 0 here
    int gemm_blocks = (gemm_waves + 7) / 8;
    gat_gemm_wmma<<<gemm_blocks, 256, 0, stream>>>(X, W, H, N, relu_in);

    int att_blocks = (N + 7) / 8;                             // 8 waves / block
    gat_att<<<att_blocks, 256, 0, stream>>>(H, att_s_w, att_d_w, a_s, a_d, N);

    int nb = (N + 255) / 256;
    gat_node_init<<<nb, 256, 0, stream>>>(a_s, a_d, eself, m, N);

    int eb = (E + 255) / 256;
    gat_edge_max<<<eb, 256, 0, stream>>>(src, dst, a_s, a_d, m, E);

    gat_node_sum_init<<<nb, 256, 0, stream>>>(eself, m, ssum, N);

    gat_edge_sum<<<eb, 256, 0, stream>>>(src, dst, a_s, a_d, m, ssum, E);

    long total = (long)N * D;
    int ob = (int)((total + 255) / 256);
    gat_out_init<<<ob, 256, 0, stream>>>(H, eself, m, ssum, bias, out, total);

    int ab = (E + 7) / 8;                                     // wave per edge
    gat_edge_agg<<<ab, 256, 0, stream>>>(H, src, dst, a_s, a_d, m, ssum, out, E);
}

extern "C" void kernel_launch(void* const* d_in, const int* in_sizes, int n_in,
                              void* d_out, int out_size, void* d_ws, size_t ws_size,
                              hipStream_t stream) {
    const float* x        = (const float*)d_in[0];
    const int*   src      = (const int*)d_in[1];
    const int*   dst      = (const int*)d_in[2];
    const float* W1       = (const float*)d_in[3];
    const float* att_src1 = (const float*)d_in[4];
    const float* att_dst1 = (const float*)d_in[5];
    const float* b1       = (const float*)d_in[6];
    const float* W2       = (const float*)d_in[7];
    const float* att_src2 = (const float*)d_in[8];
    const float* att_dst2 = (const float*)d_in[9];
    const float* b2       = (const float*)d_in[10];

    int N = in_sizes[0] / D;
    int E = in_sizes[1];

    // Workspace layout: h[N*128] | a_s[N] | a_d[N] | m[N] | ssum[N] | eself[N]
    float* ws    = (float*)d_ws;
    float* h     = ws;
    float* a_s   = h + (size_t)N * D;
    float* a_d   = a_s + N;
    float* m     = a_d + N;
    float* ssum  = m + N;
    float* eself = ssum + N;

    float* out = (float*)d_out;

    // Layer 1: x -> out (d_out temporarily holds relu-input of layer 2;
    // relu fused into layer-2 GEMM A-load).
    run_layer(x, W1, att_src1, att_dst1, b1, src, dst,
              h, a_s, a_d, m, ssum, eself, out, N, E, /*relu_in=*/0, stream);

    // Layer 2: relu(out) -> h -> ... -> out (stream-ordered, so reusing d_out
    // as both GEMM input and final aggregation target is safe).
    run_layer(out, W2, att_src2, att_dst2, b2, src, dst,
              h, a_s, a_d, m, ssum, eself, out, N, E, /*relu_in=*/1, stream);
}